// Encoder_18880676233357
// MI455X (gfx1250) — compile-verified
//
#include <hip/hip_runtime.h>
#include <hip/hip_bf16.h>

typedef __attribute__((ext_vector_type(2))) float v2f;
typedef __attribute__((ext_vector_type(8))) float v8f;

#define HID 128
#define AS_STRIDE 132   // LDS row pitch in floats; 132 % 64 = 4 -> conflict-free column reads
#define TILE_M 32       // rows per block (two 16-row WMMA tiles sharing B fragments)

// ---------------------------------------------------------------------------
// zero fill
// ---------------------------------------------------------------------------
__global__ void zero_kernel(float* __restrict__ p, int n) {
    int i = blockIdx.x * blockDim.x + threadIdx.x;
    if (i < n) p[i] = 0.0f;
}

// ---------------------------------------------------------------------------
// GINE edge message + scatter: agg[dst] += relu(z[src] + edge_attr)
// one wave per edge (32 lanes x float4 = 128 floats); coalesced 512B loads,
// 4 non-returning fp32 global atomics per lane
// ---------------------------------------------------------------------------
__global__ void edge_scatter_kernel(const float* __restrict__ z,
                                    const long long* __restrict__ ei,
                                    const float* __restrict__ ea,
                                    float* agg, int E) {
    int gt = blockIdx.x * blockDim.x + threadIdx.x;
    if (gt >= E * 32) return;
    int e = gt >> 5;
    int c = (gt & 31) * 4;
    int src = (int)ei[e];
    int dst = (int)ei[E + e];
    const float4 zv = *(const float4*)(z + (size_t)src * HID + c);
    const float4 ev = *(const float4*)(ea + (size_t)e * HID + c);
    float4 m;
    m.x = fmaxf(zv.x + ev.x, 0.0f);
    m.y = fmaxf(zv.y + ev.y, 0.0f);
    m.z = fmaxf(zv.z + ev.z, 0.0f);
    m.w = fmaxf(zv.w + ev.w, 0.0f);
    float* ap = agg + (size_t)dst * HID + c;
    atomicAdd(ap + 0, m.x);
    atomicAdd(ap + 1, m.y);
    atomicAdd(ap + 2, m.z);
    atomicAdd(ap + 3, m.w);
}

// ---------------------------------------------------------------------------
// C[M x 128] = prep(A)[M x 128] @ B[128 x 128] + bias, optional output relu.
//   mode 0: prep(A) = A0 + A1                        (residual + aggregation)
//   mode 1: prep(A) = relu(A0*scale[col]+shift[col]) (fused BN + ReLU)
// Block = 256 threads = 8 waves; block owns 32 output rows (two 16-row WMMA
// tiles); wave w owns output cols [16w, 16w+16). A staged in LDS; each K-step
// loads one B fragment and feeds TWO V_WMMA_F32_16X16X4_F32 (B reuse).
// C may alias A0/A1: A is fully staged to LDS before any store, and each
// block touches only its own rows.
// ---------------------------------------------------------------------------
__global__ __launch_bounds__(256)
void gemm_wmma_kernel(const float* A0, const float* A1,
                      const float* __restrict__ scaleB,
                      const float* __restrict__ shiftB, int mode,
                      const float* __restrict__ B,
                      const float* __restrict__ bias,
                      float* C, int M, int reluOut) {
    __shared__ float As[TILE_M * AS_STRIDE];

    const int row0 = blockIdx.x * TILE_M;
    const int t = threadIdx.x;
    const bool fullTile = (row0 + TILE_M <= M);

    // stage A tile (32 x 128): each thread loads 16 floats, applying prep()
    {
        const int r  = t >> 4;          // 0..15
        const int c0 = (t & 15) * 8;    // 0,8,...,120
        #pragma unroll
        for (int half = 0; half < 2; ++half) {
            const int rr = r + 16 * half;
            const int gr = row0 + rr;
            const bool ok = fullTile || (gr < M);
            #pragma unroll
            for (int j = 0; j < 8; ++j) {
                const int col = c0 + j;
                float v = 0.0f;
                if (ok) {
                    const float a = A0[(size_t)gr * HID + col];
                    if (mode == 0) {
                        v = a + A1[(size_t)gr * HID + col];
                    } else {
                        v = fmaf(a, scaleB[col], shiftB[col]);
                        v = fmaxf(v, 0.0f);
                    }
                }
                As[rr * AS_STRIDE + col] = v;
            }
        }
    }
    __syncthreads();

    const int wave = t >> 5;        // 0..7 -> N slice
    const int lane = t & 31;
    const int hi   = lane >> 4;     // 0: K pair {0,1}; 1: K pair {2,3}
    const int lo   = lane & 15;
    const int col  = wave * 16 + lo;

    v8f acc0, acc1;
    {
        const float bv = bias[col];
        #pragma unroll
        for (int v = 0; v < 8; ++v) { acc0[v] = bv; acc1[v] = bv; }
    }

    // K loop: 128 / 4 = 32 steps, 2 WMMAs per step (shared B fragment)
    #pragma unroll 4
    for (int k0 = 0; k0 < HID; k0 += 4) {
        const int ka = k0 + 2 * hi;
        v2f b;
        b.x = B[(size_t)ka * HID + col];
        b.y = B[(size_t)(ka + 1) * HID + col];
        v2f a0, a1;
        a0.x = As[lo * AS_STRIDE + ka];
        a0.y = As[lo * AS_STRIDE + ka + 1];
        a1.x = As[(16 + lo) * AS_STRIDE + ka];
        a1.y = As[(16 + lo) * AS_STRIDE + ka + 1];
        acc0 = __builtin_amdgcn_wmma_f32_16x16x4_f32(
            false, a0, false, b, (short)0, acc0, false, false);
        acc1 = __builtin_amdgcn_wmma_f32_16x16x4_f32(
            false, a1, false, b, (short)0, acc1, false, false);
    }

    // C/D layout: VGPR v -> row (v + 8*hi), col = lane&15 within wave slice
    if (fullTile) {
        #pragma unroll
        for (int v = 0; v < 8; ++v) {
            const int gr = row0 + v + 8 * hi;
            float o0 = acc0[v];
            float o1 = acc1[v];
            if (reluOut) { o0 = fmaxf(o0, 0.0f); o1 = fmaxf(o1, 0.0f); }
            C[(size_t)gr * HID + col] = o0;
            C[(size_t)(gr + 16) * HID + col] = o1;
        }
    } else {
        #pragma unroll
        for (int v = 0; v < 8; ++v) {
            const int gr = row0 + v + 8 * hi;
            float o0 = acc0[v];
            float o1 = acc1[v];
            if (reluOut) { o0 = fmaxf(o0, 0.0f); o1 = fmaxf(o1, 0.0f); }
            if (gr < M)      C[(size_t)gr * HID + col] = o0;
            if (gr + 16 < M) C[(size_t)(gr + 16) * HID + col] = o1;
        }
    }
}

// ---------------------------------------------------------------------------
// per-column sum / sum-of-squares (128 threads = 128 columns, 256 rows/block)
// ---------------------------------------------------------------------------
__global__ void colstats_kernel(const float* __restrict__ X, int M,
                                float* __restrict__ sum, float* __restrict__ sq) {
    const int col = threadIdx.x;              // 0..127
    const int r0 = blockIdx.x * 256;
    int rend = r0 + 256; if (rend > M) rend = M;
    float s = 0.0f, q = 0.0f;
    for (int r = r0; r < rend; ++r) {
        const float v = X[(size_t)r * HID + col];
        s += v;
        q = fmaf(v, v, q);
    }
    atomicAdd(&sum[col], s);
    atomicAdd(&sq[col], q);
}

// ---------------------------------------------------------------------------
// BN(train, biased var) -> per-column scale/shift
// ---------------------------------------------------------------------------
__global__ void bnfinalize_kernel(const float* __restrict__ sum,
                                  const float* __restrict__ sq,
                                  const float* __restrict__ g,
                                  const float* __restrict__ be, int M,
                                  float* __restrict__ scale,
                                  float* __restrict__ shift) {
    const int c = threadIdx.x;
    const float invM = 1.0f / (float)M;
    const float mean = sum[c] * invM;
    const float var  = sq[c] * invM - mean * mean;
    const float sc   = g[c] * rsqrtf(var + 1e-5f);
    scale[c] = sc;
    shift[c] = be[c] - mean * sc;
}

// ---------------------------------------------------------------------------
// elementwise X = X*scale[col] + shift[col]
// ---------------------------------------------------------------------------
__global__ void bnapply_kernel(float* __restrict__ X,
                               const float* __restrict__ scale,
                               const float* __restrict__ shift, int n) {
    int i = blockIdx.x * blockDim.x + threadIdx.x;
    if (i < n) {
        const int c = i & (HID - 1);
        X[i] = fmaf(X[i], scale[c], shift[c]);
    }
}

// ---------------------------------------------------------------------------
extern "C" void kernel_launch(void* const* d_in, const int* in_sizes, int n_in,
                              void* d_out, int out_size, void* d_ws, size_t ws_size,
                              hipStream_t stream) {
    const float*     x   = (const float*)d_in[0];
    const long long* ei  = (const long long*)d_in[1];
    const float*     ea  = (const float*)d_in[2];
    const float*     W1  = (const float*)d_in[3];
    const float*     b1  = (const float*)d_in[4];
    const float*     g1  = (const float*)d_in[5];
    const float*     be1 = (const float*)d_in[6];
    const float*     W2  = (const float*)d_in[7];
    const float*     b2  = (const float*)d_in[8];
    const float*     bng = (const float*)d_in[9];
    const float*     bnb = (const float*)d_in[10];
    float* out = (float*)d_out;

    const int M = in_sizes[0] / HID;      // 50000 nodes
    const int E = in_sizes[1] / 2;        // 640000 edges
    const int NF = M * HID;               // floats per node-feature buffer

    // workspace layout (floats): buf0 (agg, then t1 in-place), zbuf, stats
    float* buf0 = (float*)d_ws;
    float* zbuf = buf0 + NF;
    float* sumb = zbuf + NF;
    float* sqb  = sumb + HID;
    float* scb  = sqb + HID;
    float* shb  = scb + HID;

    const int zeroGridNF = (NF + 255) / 256;
    const int scatGrid   = (E * 32 + 255) / 256;
    const int gemmGrid   = (M + TILE_M - 1) / TILE_M;
    const int statGrid   = (M + 255) / 256;

    const float* zcur = x;
    const int NLAYERS = 4;
    for (int i = 0; i < NLAYERS; ++i) {
        const float* W1i = W1 + (size_t)i * HID * HID;
        const float* W2i = W2 + (size_t)i * HID * HID;

        // agg = 0 ; stats = 0
        zero_kernel<<<zeroGridNF, 256, 0, stream>>>(buf0, NF);
        zero_kernel<<<1, 256, 0, stream>>>(sumb, 2 * HID);

        // agg += relu(z[src] + e)  scattered to dst
        edge_scatter_kernel<<<scatGrid, 256, 0, stream>>>(zcur, ei, ea, buf0, E);

        // t1 = (z + agg) @ W1 + b1   (in-place over buf0; safe: A staged to LDS)
        gemm_wmma_kernel<<<gemmGrid, 256, 0, stream>>>(
            zcur, buf0, nullptr, nullptr, /*mode=*/0,
            W1i, b1 + i * HID, buf0, M, /*reluOut=*/0);

        // BN1 stats over t1 columns -> scale/shift
        colstats_kernel<<<statGrid, HID, 0, stream>>>(buf0, M, sumb, sqb);
        bnfinalize_kernel<<<1, HID, 0, stream>>>(sumb, sqb, g1 + i * HID,
                                                 be1 + i * HID, M, scb, shb);

        // z = relu( relu(BN(t1)) @ W2 + b2 )
        float* outz = (i == NLAYERS - 1) ? out : zbuf;
        gemm_wmma_kernel<<<gemmGrid, 256, 0, stream>>>(
            buf0, nullptr, scb, shb, /*mode=*/1,
            W2i, b2 + i * HID, outz, M, /*reluOut=*/1);

        if (i != NLAYERS - 1) {
            // inter-layer BN on z
            zero_kernel<<<1, 256, 0, stream>>>(sumb, 2 * HID);
            colstats_kernel<<<statGrid, HID, 0, stream>>>(zbuf, M, sumb, sqb);
            bnfinalize_kernel<<<1, HID, 0, stream>>>(sumb, sqb, bng + i * HID,
                                                     bnb + i * HID, M, scb, shb);
            bnapply_kernel<<<zeroGridNF, 256, 0, stream>>>(zbuf, scb, shb, NF);
            zcur = zbuf;
        }
    }
}